// SRModel_49658411877054
// MI455X (gfx1250) — compile-verified
//
#include <hip/hip_runtime.h>
#include <hip/hip_bf16.h>
#include <cstdint>

typedef __attribute__((ext_vector_type(2))) float v2f;
typedef __attribute__((ext_vector_type(4))) float f32x4;
typedef __attribute__((ext_vector_type(8))) float v8f;
typedef int v4i_g __attribute__((vector_size(16)));   // matches builtin param type

#define ASG __attribute__((address_space(1)))
#define ASL __attribute__((address_space(3)))

#if defined(__HIP_DEVICE_COMPILE__) && __has_builtin(__builtin_amdgcn_global_load_async_to_lds_b128)
#define HAVE_ASYNC_COPY 1
#else
#define HAVE_ASYNC_COPY 0
#endif

// ---------------- model dims (fixed by reference) ----------------
constexpr int NB   = 16;    // batch
constexpr int F0   = 128;   // input freq
constexpr int NT   = 1024;  // input time
constexpr int NC   = 32;    // conv channels
constexpr int FR   = 64;    // freq after stride-2
constexpr int LSEQ = 512;   // time after stride-2
constexpr int DD   = 2048;  // NC*FR
constexpr int HH   = 1024;  // SRU hidden per direction
constexpr int KOUT = 6144;  // 3*2*HH
constexpr int NV   = 29;    // vocab

// ---------------- stem conv: (16,1,128,1024) -> (16,32,64,512), 3x3 s2 p1 --
__global__ __launch_bounds__(256) void k_conv_stem(
    const float* __restrict__ x, const float* __restrict__ w,
    const float* __restrict__ bias, float* __restrict__ y)
{
  int idx = blockIdx.x * 256 + threadIdx.x;          // 2^24 threads
  int t = idx & (LSEQ - 1);
  int f = (idx >> 9) & (FR - 1);
  int c = (idx >> 15) & (NC - 1);
  int b = idx >> 20;
  float acc = bias[c];
#pragma unroll
  for (int kf = 0; kf < 3; ++kf) {
    int fi = 2 * f + kf - 1;
    if ((unsigned)fi >= (unsigned)F0) continue;
#pragma unroll
    for (int kt = 0; kt < 3; ++kt) {
      int ti = 2 * t + kt - 1;
      if ((unsigned)ti >= (unsigned)NT) continue;
      acc += x[((size_t)b * F0 + fi) * NT + ti] * w[c * 9 + kf * 3 + kt];
    }
  }
  y[(((size_t)b * NC + c) * FR + f) * LSEQ + t] = acc;
}

// ---------------- residual conv 3x3 s1 p1 + BN (+res) + ReLU --------------
__global__ __launch_bounds__(256) void k_conv_res(
    const float* __restrict__ x, const float* __restrict__ w,
    const float* __restrict__ bias,
    const float* __restrict__ g, const float* __restrict__ bb,
    const float* __restrict__ bm, const float* __restrict__ bv,
    const float* __restrict__ res, float* __restrict__ y, int add_res)
{
  int idx = blockIdx.x * 256 + threadIdx.x;
  int t  = idx & (LSEQ - 1);
  int f  = (idx >> 9) & (FR - 1);
  int co = (idx >> 15) & (NC - 1);
  int b  = idx >> 20;
  float acc = bias[co];
  for (int ci = 0; ci < NC; ++ci) {
    const float* xc = x + ((size_t)b * NC + ci) * FR * LSEQ;
    const float* wc = w + ((size_t)co * NC + ci) * 9;
#pragma unroll
    for (int kf = 0; kf < 3; ++kf) {
      int fi = f + kf - 1;
      if ((unsigned)fi >= (unsigned)FR) continue;
#pragma unroll
      for (int kt = 0; kt < 3; ++kt) {
        int ti = t + kt - 1;
        if ((unsigned)ti >= (unsigned)LSEQ) continue;
        acc += xc[(size_t)fi * LSEQ + ti] * wc[kf * 3 + kt];
      }
    }
  }
  float z = (acc - bm[co]) * rsqrtf(bv[co] + 1e-5f) * g[co] + bb[co];
  size_t oi = (((size_t)b * NC + co) * FR + f) * LSEQ + t;
  if (add_res) z += res[oi];
  y[oi] = fmaxf(z, 0.f);
}

// ---------------- (B,C,FR,L) -> (L,B,D) with D = c*FR + f ----------------
__global__ __launch_bounds__(256) void k_to_lbd(
    const float* __restrict__ h, float* __restrict__ y)
{
  int idx = blockIdx.x * 256 + threadIdx.x;          // 2^24, output-linear
  int dd = idx & (DD - 1);
  int b  = (idx >> 11) & (NB - 1);
  int t  = idx >> 15;
  int c = dd >> 6;
  int f = dd & 63;
  y[idx] = h[(((size_t)b * NC + c) * FR + f) * LSEQ + t];
}

// ---------------- generic row LayerNorm (in-place capable) ---------------
__global__ __launch_bounds__(256) void k_layernorm(
    const float* __restrict__ x, const float* __restrict__ w,
    const float* __restrict__ b, float* __restrict__ y, int D)
{
  __shared__ float red[256];
  __shared__ float red2[256];
  int row = blockIdx.x;
  const float* xr = x + (size_t)row * D;
  float s = 0.f, s2 = 0.f;
  for (int i = threadIdx.x; i < D; i += 256) { float v = xr[i]; s += v; s2 += v * v; }
  red[threadIdx.x] = s; red2[threadIdx.x] = s2;
  __syncthreads();
  for (int o = 128; o > 0; o >>= 1) {
    if (threadIdx.x < o) { red[threadIdx.x] += red[threadIdx.x + o];
                           red2[threadIdx.x] += red2[threadIdx.x + o]; }
    __syncthreads();
  }
  float mean = red[0] / D;
  float rstd = rsqrtf(red2[0] / D - mean * mean + 1e-5f);
  float* yr = y + (size_t)row * D;
  for (int i = threadIdx.x; i < D; i += 256)
    yr[i] = (xr[i] - mean) * rstd * w[i] + b[i];
}

// ---------------- per-row mean/rstd (feeds LN-fused GEMM) ----------------
__global__ __launch_bounds__(256) void k_rowstats(
    const float* __restrict__ x, float* __restrict__ stats, int D)
{
  __shared__ float red[256];
  __shared__ float red2[256];
  int row = blockIdx.x;
  const float* xr = x + (size_t)row * D;
  float s = 0.f, s2 = 0.f;
  for (int i = threadIdx.x; i < D; i += 256) { float v = xr[i]; s += v; s2 += v * v; }
  red[threadIdx.x] = s; red2[threadIdx.x] = s2;
  __syncthreads();
  for (int o = 128; o > 0; o >>= 1) {
    if (threadIdx.x < o) { red[threadIdx.x] += red[threadIdx.x + o];
                           red2[threadIdx.x] += red2[threadIdx.x + o]; }
    __syncthreads();
  }
  if (threadIdx.x == 0) {
    float mean = red[0] / D;
    stats[2 * row]     = mean;
    stats[2 * row + 1] = rsqrtf(red2[0] / D - mean * mean + 1e-5f);
  }
}

// ---------------- LN-fused f32 WMMA GEMM: U = LN(X) @ W ------------------
// X:(M,K) row-major, W:(K,N) row-major, U:(M,N). M%64==0, N%256==0, K%32==0.
// Block 64x256 (8 waves, 2x4), wave tile 32x64, double-buffered LDS,
// B tile staged through GLOBAL_LOAD_ASYNC_TO_LDS_B128 when available.
constexpr int TBM = 64, TBN = 256, TBK = 32;

__global__ __launch_bounds__(256) void k_gemm_ln_wmma(
    const float* __restrict__ X, const float* __restrict__ stats,
    const float* __restrict__ lnw, const float* __restrict__ lnb,
    const float* __restrict__ W, float* __restrict__ U,
    int M, int N, int K)
{
  __shared__ float As[2][TBK][TBM + 1];     // k-major (transposed), 2x8.1KB
  __shared__ float Bs[2][TBK][TBN + 4];     // 2x33.3KB, rows 16B-aligned

  const int tid   = threadIdx.x;
  const int lane  = tid & 31;
  const int wave  = tid >> 5;               // 8 waves
  const int waveM = wave & 1;               // 2 rows of waves
  const int waveN = wave >> 1;              // 4 cols of waves
  const int m0 = blockIdx.x * TBM;
  const int n0 = blockIdx.y * TBN;
  const int half = lane >> 4;               // K-pair select per ISA A/B layout
  const int mr = lane & 15;
  const int nr = lane & 15;
  const int mW = waveM * 32;
  const int nW = waveN * 64;

  v8f acc[2][4] = {};                       // eight 16x16 f32 tiles per wave

  auto loadA = [&](int k0, int buf) {
#pragma unroll
    for (int e = tid; e < TBM * TBK; e += 256) {
      int m = e >> 5, k = e & 31;
      int gm = m0 + m;
      float mean = stats[2 * gm], rstd = stats[2 * gm + 1];
      float v = X[(size_t)gm * K + k0 + k];
      As[buf][k][m] = (v - mean) * rstd * lnw[k0 + k] + lnb[k0 + k];
    }
  };
  auto loadB = [&](int k0, int buf) {
#pragma unroll
    for (int e = tid; e < (TBK * TBN) / 4; e += 256) {
      int k = e >> 6, n4 = e & 63;
      const float* src = W + (size_t)(k0 + k) * N + n0 + n4 * 4;
      float* dst = &Bs[buf][k][n4 * 4];
#if HAVE_ASYNC_COPY
      __builtin_amdgcn_global_load_async_to_lds_b128(
          (ASG v4i_g*)src, (ASL v4i_g*)dst, 0, 0);
#else
      *(f32x4*)dst = *(const f32x4*)src;
#endif
    }
  };
  auto waitAsync = [&]() {
#if HAVE_ASYNC_COPY
#if __has_builtin(__builtin_amdgcn_s_wait_asynccnt)
    __builtin_amdgcn_s_wait_asynccnt(0);
#else
    asm volatile("s_wait_asynccnt 0" ::: "memory");
#endif
#endif
  };

  loadA(0, 0);
  loadB(0, 0);
  waitAsync();
  __syncthreads();

  const int nChunk = K / TBK;
  for (int c = 0; c < nChunk; ++c) {
    const int buf = c & 1;
    if (c + 1 < nChunk) {                   // prefetch next chunk into buf^1
      int k0n = (c + 1) * TBK;
      loadA(k0n, buf ^ 1);
      loadB(k0n, buf ^ 1);
      __builtin_prefetch(&X[(size_t)(m0 + wave * 8) * K + k0n], 0, 0);
    }
#pragma unroll
    for (int kk = 0; kk < TBK; kk += 4) {
      // A frag 16x4: vgpr0 = K {0|2}, vgpr1 = K {1|3} (wave halves)
      v2f a[2];
#pragma unroll
      for (int mi = 0; mi < 2; ++mi) {
        a[mi].x = As[buf][kk + 2 * half + 0][mW + mi * 16 + mr];
        a[mi].y = As[buf][kk + 2 * half + 1][mW + mi * 16 + mr];
      }
#pragma unroll
      for (int nt = 0; nt < 4; ++nt) {
        v2f bfv;
        bfv.x = Bs[buf][kk + 2 * half + 0][nW + nt * 16 + nr];
        bfv.y = Bs[buf][kk + 2 * half + 1][nW + nt * 16 + nr];
#pragma unroll
        for (int mi = 0; mi < 2; ++mi)
          acc[mi][nt] = __builtin_amdgcn_wmma_f32_16x16x4_f32(
              false, a[mi], false, bfv, (short)0, acc[mi][nt], false, false);
      }
    }
    waitAsync();                            // next-chunk async B in flight
    __syncthreads();
  }

  // C/D layout: vgpr r -> rows (r, r+8); lanes 0-15 / 16-31 select row half
#pragma unroll
  for (int mi = 0; mi < 2; ++mi)
#pragma unroll
    for (int nt = 0; nt < 4; ++nt)
#pragma unroll
      for (int r = 0; r < 8; ++r) {
        int row = m0 + mW + mi * 16 + r + half * 8;
        int col = n0 + nW + nt * 16 + nr;
        U[(size_t)row * N + col] = acc[mi][nt][r];
      }
}

// ---------------- bidirectional SRU recurrence ---------------------------
__global__ __launch_bounds__(256) void k_sru_scan(
    const float* __restrict__ U, const float* __restrict__ xprev,
    const float* __restrict__ vf, const float* __restrict__ vr,
    const float* __restrict__ bf, const float* __restrict__ br,
    float* __restrict__ out)
{
  int g = blockIdx.x * 256 + threadIdx.x;   // 2*NB*HH = 32768
  int h = g & (HH - 1);
  int b = (g >> 10) & (NB - 1);
  int d = g >> 14;
  float VF = vf[d * HH + h], VR = vr[d * HH + h];
  float BF = bf[d * HH + h], BR = br[d * HH + h];
  float c = 0.f;
  for (int i = 0; i < LSEQ; ++i) {
    int t = d ? (LSEQ - 1 - i) : i;
    size_t base = (size_t)t * NB + b;
    size_t ub = (base * 2 + d) * (size_t)(3 * HH) + h;
    float xt = U[ub];
    float fp = U[ub + HH];
    float rp = U[ub + 2 * HH];
    float sk = xprev[(base * 2 + d) * HH + h];
    float f = 1.f / (1.f + __expf(-(fp + VF * c + BF)));
    float r = 1.f / (1.f + __expf(-(rp + VR * c + BR)));
    c = f * c + (1.f - f) * xt;
    out[(base * 2 + d) * HH + h] = r * c + (1.f - r) * sk;
  }
}

// ---------------- dir-sum + LN + (1024 x 29) classifier ------------------
__global__ __launch_bounds__(256) void k_classifier(
    const float* __restrict__ x, const float* __restrict__ lnw,
    const float* __restrict__ lnb, const float* __restrict__ cw,
    float* __restrict__ out)
{
  __shared__ float sh[HH];
  __shared__ float red[256];
  __shared__ float red2[256];
  int row = blockIdx.x;                     // t*NB + b
  int t = row / NB, b = row % NB;
  const float* xr = x + (size_t)row * (2 * HH);
  float s = 0.f, s2 = 0.f;
  for (int i = threadIdx.x; i < HH; i += 256) {
    float v = xr[i] + xr[i + HH];
    sh[i] = v; s += v; s2 += v * v;
  }
  red[threadIdx.x] = s; red2[threadIdx.x] = s2;
  __syncthreads();
  for (int o = 128; o > 0; o >>= 1) {
    if (threadIdx.x < o) { red[threadIdx.x] += red[threadIdx.x + o];
                           red2[threadIdx.x] += red2[threadIdx.x + o]; }
    __syncthreads();
  }
  float mean = red[0] / HH;
  float rstd = rsqrtf(red2[0] / HH - mean * mean + 1e-5f);
  for (int i = threadIdx.x; i < HH; i += 256)
    sh[i] = (sh[i] - mean) * rstd * lnw[i] + lnb[i];
  __syncthreads();
  if (threadIdx.x < NV) {
    float acc = 0.f;
    for (int hh = 0; hh < HH; ++hh) acc += sh[hh] * cw[hh * NV + threadIdx.x];
    out[((size_t)b * LSEQ + t) * NV + threadIdx.x] = acc;
  }
}

// ---------------- host orchestration -------------------------------------
extern "C" void kernel_launch(void* const* d_in, const int* in_sizes, int n_in,
                              void* d_out, int out_size, void* d_ws, size_t ws_size,
                              hipStream_t stream)
{
  const float* x     = (const float*)d_in[0];
  const float* cnnw  = (const float*)d_in[1];
  const float* cnnb  = (const float*)d_in[2];
  const float* rc1w  = (const float*)d_in[3];
  const float* rc1b  = (const float*)d_in[4];
  const float* rc2w  = (const float*)d_in[5];
  const float* rc2b  = (const float*)d_in[6];
  const float* bn1g  = (const float*)d_in[7];
  const float* bn1b  = (const float*)d_in[8];
  const float* bn1m  = (const float*)d_in[9];
  const float* bn1v  = (const float*)d_in[10];
  const float* bn2g  = (const float*)d_in[11];
  const float* bn2b  = (const float*)d_in[12];
  const float* bn2m  = (const float*)d_in[13];
  const float* bn2v  = (const float*)d_in[14];
  const float* flnw  = (const float*)d_in[15];
  const float* flnb  = (const float*)d_in[16];
  const float* sruW  = (const float*)d_in[17];
  const float* sruvf = (const float*)d_in[18];
  const float* sruvr = (const float*)d_in[19];
  const float* srubf = (const float*)d_in[20];
  const float* srubr = (const float*)d_in[21];
  const float* slnw  = (const float*)d_in[22];
  const float* slnb  = (const float*)d_in[23];
  const float* clnw  = (const float*)d_in[24];
  const float* clnb  = (const float*)d_in[25];
  const float* clsw  = (const float*)d_in[26];
  float* out = (float*)d_out;

  const size_t actElems = (size_t)NB * NC * FR * LSEQ;   // 16.7M
  const size_t lbdElems = (size_t)LSEQ * NB * DD;        // 16.7M
  const size_t uElems   = (size_t)LSEQ * NB * KOUT;      // 50.3M

  // ws layout: [U (192MB) overlays conv ping-pong r0|r1][xA 64MB][xB 64MB][stats]
  float* r0 = (float*)d_ws;
  float* r1 = r0 + actElems;
  float* U  = r0;
  float* xA = r0 + uElems;
  float* xB = xA + lbdElems;
  float* stats = xB + lbdElems;

  dim3 blk(256);
  const int nAct = (int)(actElems / 256);

  // stem + residual stack (conv2 fused BN+res+ReLU writes r0 in place)
  k_conv_stem<<<nAct, blk, 0, stream>>>(x, cnnw, cnnb, r0);
  for (int i = 0; i < 3; ++i) {
    k_conv_res<<<nAct, blk, 0, stream>>>(r0, rc1w + (size_t)i * NC * NC * 9,
        rc1b + i * NC, bn1g + i * NC, bn1b + i * NC, bn1m + i * NC, bn1v + i * NC,
        nullptr, r1, 0);
    k_conv_res<<<nAct, blk, 0, stream>>>(r1, rc2w + (size_t)i * NC * NC * 9,
        rc2b + i * NC, bn2g + i * NC, bn2b + i * NC, bn2m + i * NC, bn2v + i * NC,
        r0, r0, 1);
  }

  // to (L,B,D) + feature LayerNorm (in place)
  k_to_lbd<<<(int)(lbdElems / 256), blk, 0, stream>>>(r0, xA);
  k_layernorm<<<LSEQ * NB, blk, 0, stream>>>(xA, flnw, flnb, xA, DD);

  // 5 SRU layers
  float* pcur = xA;
  float* pnext = xB;
  for (int i = 0; i < 5; ++i) {
    k_rowstats<<<LSEQ * NB, blk, 0, stream>>>(pcur, stats, DD);
    dim3 grid((LSEQ * NB) / TBM, KOUT / TBN);
    k_gemm_ln_wmma<<<grid, blk, 0, stream>>>(pcur, stats,
        slnw + (size_t)i * DD, slnb + (size_t)i * DD,
        sruW + (size_t)i * DD * KOUT, U, LSEQ * NB, KOUT, DD);
    k_sru_scan<<<(2 * NB * HH) / 256, blk, 0, stream>>>(U, pcur,
        sruvf + (size_t)i * 2 * HH, sruvr + (size_t)i * 2 * HH,
        srubf + (size_t)i * 2 * HH, srubr + (size_t)i * 2 * HH, pnext);
    float* tmp = pcur; pcur = pnext; pnext = tmp;
  }

  // sum dirs + LN + classifier -> (B, L, 29)
  k_classifier<<<LSEQ * NB, blk, 0, stream>>>(pcur, clnw, clnb, clsw, out);
}